// final_row_model_90494960926732
// MI455X (gfx1250) — compile-verified
//
#include <hip/hip_runtime.h>

typedef float v2f __attribute__((ext_vector_type(2)));
typedef float v8f __attribute__((ext_vector_type(8)));

#define B_ 128
#define M_ 10000
#define P_ 64
#define W_ 24

__global__ __launch_bounds__(256)
void chanlin_wmma_f32_kernel(const float* __restrict__ x,
                             const float* __restrict__ w,
                             const float* __restrict__ bias,
                             float* __restrict__ out)
{
    const int j    = blockIdx.x;           // channel index (0..M_-1)
    const int lane = threadIdx.x & 31;
    const int wave = threadIdx.x >> 5;     // 0..7

    const int l15   = lane & 15;
    const int half  = lane >> 4;           // 0 or 1 (upper half-wave)
    const int mBase = wave * 16;           // batch-row tile base

    const float* xj = x + (size_t)j * P_;          // + m * (M_*P_) per row
    const float* wj = w + (size_t)j * (P_ * W_);   // weight[j] : [64][24]

    v8f c0 = {0.f,0.f,0.f,0.f,0.f,0.f,0.f,0.f};   // N-tile 0 (cols 0..15)
    v8f c1 = {0.f,0.f,0.f,0.f,0.f,0.f,0.f,0.f};   // N-tile 1 (cols 16..23, padded)

    // A-fragment row for this lane (16x4 f32 layout: lanes 0-15 = M 0..15)
    const int m = mBase + l15;
    const float* xrow = xj + (size_t)m * (M_ * P_);

    const int  n0   = l15;                 // tile-0 output column (w)
    const int  n1   = 16 + l15;            // tile-1 output column
    const bool n1ok = (n1 < W_);
    const int  n1c  = n1ok ? n1 : (W_ - 1);  // clamped addr; value zeroed below

#pragma unroll
    for (int k0 = 0; k0 < P_; k0 += 4) {
        const int kb = k0 + 2 * half;      // K-pair base for this half-wave

        // A 16x4 fragment: contiguous b64 load (x row contiguous in P)
        v2f a = *(const v2f*)(xrow + kb);

        // B 4x16 fragments: lane holds column n, VGPR0/1 hold K=kb/kb+1
        v2f b0, b1;
        b0.x = wj[(size_t)kb * W_ + n0];
        b0.y = wj[(size_t)(kb + 1) * W_ + n0];
        float t0 = wj[(size_t)kb * W_ + n1c];
        float t1 = wj[(size_t)(kb + 1) * W_ + n1c];
        b1.x = n1ok ? t0 : 0.0f;           // zero-pad cols 24..31
        b1.y = n1ok ? t1 : 0.0f;

        // D = A*B + C   (8 args: neg_a, A, neg_b, B, c_mod, C, reuse_a, reuse_b)
        c0 = __builtin_amdgcn_wmma_f32_16x16x4_f32(false, a, false, b0,
                                                   (short)0, c0, false, false);
        c1 = __builtin_amdgcn_wmma_f32_16x16x4_f32(false, a, false, b1,
                                                   (short)0, c1, false, false);
    }

    // bias[j, n] — per-column, uniform over the 8 rows a lane holds
    const float bias0 = bias[(size_t)j * W_ + n0];
    const float bias1 = n1ok ? bias[(size_t)j * W_ + n1] : 0.0f;

    // C layout: lane L holds rows m = mBase + r + 8*(L>>4), col n = L&15
    const size_t outCol0 = (size_t)n0 * M_ + j;
    const size_t outCol1 = (size_t)n1 * M_ + j;
#pragma unroll
    for (int r = 0; r < 8; ++r) {
        const int mm = mBase + r + 8 * half;
        out[(size_t)mm * (W_ * M_) + outCol0] = c0[r] + bias0;
        if (n1ok)
            out[(size_t)mm * (W_ * M_) + outCol1] = c1[r] + bias1;
    }
}

extern "C" void kernel_launch(void* const* d_in, const int* in_sizes, int n_in,
                              void* d_out, int out_size, void* d_ws, size_t ws_size,
                              hipStream_t stream) {
    const float* x    = (const float*)d_in[0];   // [128, 10000, 64]
    const float* wgt  = (const float*)d_in[1];   // [10000, 64, 24]
    const float* bias = (const float*)d_in[2];   // [10000, 24]
    float*       out  = (float*)d_out;           // [128, 24, 10000]

    chanlin_wmma_f32_kernel<<<M_, 256, 0, stream>>>(x, wgt, bias, out);
}